// GAT_44487271252171
// MI455X (gfx1250) — compile-verified
//
#include <hip/hip_runtime.h>

#define NN 50000
#define HH 4
#define CC 64
#define FF 256   // H*C
#define BN_EPS 1e-5f

typedef float v2f __attribute__((ext_vector_type(2)));
typedef float v8f __attribute__((ext_vector_type(8)));

#if defined(__has_builtin)
#if __has_builtin(__builtin_amdgcn_wmma_f32_16x16x4_f32)
#define USE_WMMA_F32 1
#endif
#endif

// ---------------------------------------------------------------------------
// GEMM: D[M,Nc] = A[M,K] @ B[K,Nc] (+ bias).
// Block = 8 waves = 8 row strips x ONE column group (TN*16 cols).
// B column tile (K x TN*16, K<=256) is staged once into LDS (<=64KB of the
// WGP's 320KB) and shared by all 8 waves; inner loop reads B via ds_load.
// WMMA: V_WMMA_F32_16X16X4_F32 (full fp32 precision, fp32 accumulate).
// A frag: lane m=L%16 holds contiguous K-pair (0,1) lanes<16 / (2,3) lanes>=16
// (one b64 load). C/D: VGPR r, lane L -> row r+(L>=16?8:0), col L%16
// (per CDNA5 ISA 7.12.2).
// ---------------------------------------------------------------------------
template <int TN>
__global__ void gemm_ker(const float* __restrict__ A, const float* __restrict__ B,
                         const float* __restrict__ bias, float* __restrict__ D,
                         int M, int K, int Nc) {
    constexpr int tileW = TN * 16;
    __shared__ float Bs[256 * tileW];            // K_max = 256
    const int lane = threadIdx.x & 31;
    const int wave = threadIdx.x >> 5;
    const int strips = M >> 4;                   // M is a multiple of 16 (50000)
    const int colGroups = Nc / tileW;
    const int colGroup = blockIdx.x % colGroups;
    const int stripGroup = blockIdx.x / colGroups;
    const int colBase = colGroup * tileW;

    // cooperative, coalesced stage of the B column tile into LDS
    for (int idx = threadIdx.x; idx < K * tileW; idx += blockDim.x) {
        const int kk  = idx / tileW;
        const int col = idx % tileW;
        Bs[idx] = B[(size_t)kk * Nc + colBase + col];
    }
    __syncthreads();

    const int rowStrip = stripGroup * 8 + wave;  // blockDim = 256 -> 8 waves
    if (rowStrip >= strips) return;

    const int m = lane & 15;
    v8f acc[TN] = {};

#ifdef USE_WMMA_F32
    const int ka = (lane < 16) ? 0 : 2;
    const float* Arow = A + (size_t)(rowStrip * 16 + m) * K;
    for (int k0 = 0; k0 < K; k0 += 4) {
        v2f a = *(const v2f*)(Arow + k0 + ka);   // contiguous K-pair, 8B aligned
#pragma unroll
        for (int t = 0; t < TN; ++t) {
            v2f b;
            b.x = Bs[(k0 + ka)     * tileW + t * 16 + m];
            b.y = Bs[(k0 + ka + 1) * tileW + t * 16 + m];
            acc[t] = __builtin_amdgcn_wmma_f32_16x16x4_f32(
                false, a, false, b, (short)0, acc[t], false, false);
        }
    }
#else
    // scalar fallback (keeps compile alive if the f32 WMMA builtin is absent)
    const int rbase0 = (lane >= 16) ? 8 : 0;
    for (int k = 0; k < K; ++k) {
#pragma unroll
        for (int t = 0; t < TN; ++t) {
            const float bv = Bs[k * tileW + t * 16 + m];
#pragma unroll
            for (int r = 0; r < 8; ++r)
                acc[t][r] += A[(size_t)(rowStrip * 16 + rbase0 + r) * K + k] * bv;
        }
    }
#endif

    const int rbase = (lane >= 16) ? 8 : 0;
#pragma unroll
    for (int t = 0; t < TN; ++t) {
        const int n = colBase + t * 16 + m;
        const float bv = bias ? bias[n] : 0.0f;
#pragma unroll
        for (int r = 0; r < 8; ++r)
            D[(size_t)(rowStrip * 16 + rbase + r) * Nc + n] = acc[t][r] + bv;
    }
}

// ---------------------------------------------------------------------------
// Attention logits: al_s[n,h] = sum_c h[n,h,c]*a_s[h,c]; same for a_d.
// ---------------------------------------------------------------------------
__global__ void logits_ker(const float* __restrict__ h, const float* __restrict__ as_,
                           const float* __restrict__ ad_, float* __restrict__ als,
                           float* __restrict__ ald) {
    int tid = blockIdx.x * blockDim.x + threadIdx.x;   // tid = node*HH + head
    if (tid >= NN * HH) return;
    const int head = tid & (HH - 1);
    const int node = tid >> 2;
    const float* hp  = h   + (size_t)node * FF + head * CC;
    const float* asp = as_ + head * CC;
    const float* adp = ad_ + head * CC;
    float s1 = 0.f, s2 = 0.f;
#pragma unroll 8
    for (int c = 0; c < CC; ++c) { float v = hp[c]; s1 += v * asp[c]; s2 += v * adp[c]; }
    als[tid] = s1;
    ald[tid] = s2;
}

__global__ void init_ker(float* __restrict__ outb, float* __restrict__ mx,
                         float* __restrict__ sm) {
    int tid = blockIdx.x * blockDim.x + threadIdx.x;
    if (tid < NN * FF) outb[tid] = 0.f;
    if (tid < NN * HH) { mx[tid] = __uint_as_float(0xFF800000u); sm[tid] = 0.f; }
}

__device__ inline void atomicMaxF(float* addr, float val) {
    // monotone int/uint reinterpretation trick (total order preserved)
    if (val >= 0.f) atomicMax((int*)addr, __float_as_int(val));
    else            atomicMin((unsigned int*)addr, __float_as_uint(val));
}

// pass A: e = leaky_relu(al_s[src]+al_d[dst]); store; segment max over dst
__global__ void edge_logit_max(const int* __restrict__ ei, int E,
                               const float* __restrict__ als, const float* __restrict__ ald,
                               float* __restrict__ eb, float* __restrict__ mx) {
    int tid = blockIdx.x * blockDim.x + threadIdx.x;   // tid = edge*HH + head
    if (tid >= (E + NN) * HH) return;
    const int e = tid >> 2, head = tid & (HH - 1);
    int s, d;
    if (e < E) { s = ei[e]; d = ei[E + e]; } else { s = d = e - E; }
    float v = als[s * HH + head] + ald[d * HH + head];
    v = (v > 0.f) ? v : 0.2f * v;
    eb[tid] = v;
    atomicMaxF(&mx[d * HH + head], v);
}

// pass B: ex = exp(e - m[dst]); store; segment sum over dst
__global__ void edge_expsum(const int* __restrict__ ei, int E,
                            const float* __restrict__ mx, float* __restrict__ eb,
                            float* __restrict__ sm) {
    int tid = blockIdx.x * blockDim.x + threadIdx.x;
    if (tid >= (E + NN) * HH) return;
    const int e = tid >> 2, head = tid & (HH - 1);
    int d = (e < E) ? ei[E + e] : (e - E);
    float ex = __expf(eb[tid] - mx[d * HH + head]);
    eb[tid] = ex;
    atomicAdd(&sm[d * HH + head], ex);
}

// pass C: out[dst] += alpha * h[src]  (one wave per edge, 8 channels/lane)
__global__ void edge_aggregate(const int* __restrict__ ei, int E,
                               const float* __restrict__ eb, const float* __restrict__ sm,
                               const float* __restrict__ h, float* __restrict__ outb) {
    const int lane = threadIdx.x & 31;
    const int wave = threadIdx.x >> 5;
    const int e = blockIdx.x * (blockDim.x >> 5) + wave;
    if (e >= E + NN) return;
    int s, d;
    if (e < E) { s = ei[e]; d = ei[E + e]; } else { s = d = e - E; }
    const float* hs = h + (size_t)s * FF;
    // prefetch the gathered 1KB row (32 lanes x 32B stride covers all lines)
    __builtin_prefetch(hs + lane * 8, 0, 3);
    float alpha[HH];
#pragma unroll
    for (int hh = 0; hh < HH; ++hh)
        alpha[hh] = eb[e * HH + hh] / (sm[d * HH + hh] + 1e-16f);
    float* od = outb + (size_t)d * FF;
#pragma unroll
    for (int j = 0; j < FF / 32; ++j) {
        const int c = lane + 32 * j;          // coalesced across lanes
        atomicAdd(&od[c], alpha[c >> 6] * hs[c]);
    }
}

// BN stats: one block per column; mean and biased variance (jnp.var ddof=0)
__global__ void bn_stats(const float* __restrict__ x, float* __restrict__ mu,
                         float* __restrict__ var) {
    __shared__ float ssum[256], ssq[256];
    const int j = blockIdx.x;
    float s = 0.f, q = 0.f;
    for (int i = threadIdx.x; i < NN; i += blockDim.x) {
        float v = x[(size_t)i * FF + j];
        s += v; q += v * v;
    }
    ssum[threadIdx.x] = s; ssq[threadIdx.x] = q;
    __syncthreads();
    for (int off = 128; off > 0; off >>= 1) {
        if ((int)threadIdx.x < off) {
            ssum[threadIdx.x] += ssum[threadIdx.x + off];
            ssq[threadIdx.x]  += ssq[threadIdx.x + off];
        }
        __syncthreads();
    }
    if (threadIdx.x == 0) {
        float m = ssum[0] * (1.0f / NN);
        mu[j] = m;
        var[j] = ssq[0] * (1.0f / NN) - m * m;
    }
}

__global__ void bn_relu(const float* __restrict__ xin, const float* __restrict__ mu,
                        const float* __restrict__ var, const float* __restrict__ g,
                        const float* __restrict__ be, float* __restrict__ y) {
    int tid = blockIdx.x * blockDim.x + threadIdx.x;
    if (tid >= NN * FF) return;
    const int j = tid & (FF - 1);
    float v = g[j] * (xin[tid] - mu[j]) * rsqrtf(var[j] + BN_EPS) + be[j];
    y[tid] = v > 0.f ? v : 0.f;
}

// ---------------------------------------------------------------------------
extern "C" void kernel_launch(void* const* d_in, const int* in_sizes, int n_in,
                              void* d_out, int out_size, void* d_ws, size_t ws_size,
                              hipStream_t stream) {
    (void)n_in; (void)out_size; (void)ws_size;
    const float* x  = (const float*)d_in[0];
    const int* eis  = (const int*)d_in[1];
    const int* eit  = (const int*)d_in[2];
    const int  E    = in_sizes[1] / 2;

    // workspace layout (floats)
    float* ws = (float*)d_ws;
    const size_t NF = (size_t)NN * FF;
    float* feat = ws;                 // layer activations  [N,256]
    float* hbuf = ws + NF;            // h = x@W            [N,256]
    float* outb = ws + 2 * NF;        // aggregated output  [N,256]
    float* als  = ws + 3 * NF;        // [N,H]
    float* ald  = als + (size_t)NN * HH;
    float* mx   = ald + (size_t)NN * HH;
    float* sm   = mx  + (size_t)NN * HH;
    float* eb   = sm  + (size_t)NN * HH;          // per-edge logits [(E+N)*H]
    float* mu   = eb  + (size_t)(E + NN) * HH;
    float* var  = mu + FF;

    const int widx[4]        = {3, 9, 15, 21};
    const int* ei_arr[4]     = {eis, eis, eit, eit};
    const int Ks[4]          = {128, FF, FF, FF};

    const int strips = NN / 16;                       // 3125
    const int stripGroups = (strips + 7) / 8;         // 8 row strips per block
    const int gemmBlk4 = stripGroups * (FF / 64);     // TN=4: 4 col groups
    const int nhBlk   = (NN * HH + 255) / 256;
    const int nfBlk   = (NN * FF + 255) / 256;
    const int ehBlk   = ((E + NN) * HH + 255) / 256;
    const int aggBlk  = (E + NN + 7) / 8;             // 8 waves/block, wave/edge

    const float* in = x;
    for (int l = 0; l < 4; ++l) {
        const float* W   = (const float*)d_in[widx[l] + 0];
        const float* as_ = (const float*)d_in[widx[l] + 1];
        const float* ad_ = (const float*)d_in[widx[l] + 2];
        const float* g   = (const float*)d_in[widx[l] + 4];
        const float* be  = (const float*)d_in[widx[l] + 5];
        const int* ei    = ei_arr[l];

        gemm_ker<4><<<gemmBlk4, 256, 0, stream>>>(in, W, nullptr, hbuf, NN, Ks[l], FF);
        logits_ker<<<nhBlk, 256, 0, stream>>>(hbuf, as_, ad_, als, ald);
        init_ker<<<nfBlk, 256, 0, stream>>>(outb, mx, sm);
        edge_logit_max<<<ehBlk, 256, 0, stream>>>(ei, E, als, ald, eb, mx);
        edge_expsum<<<ehBlk, 256, 0, stream>>>(ei, E, mx, eb, sm);
        edge_aggregate<<<aggBlk, 256, 0, stream>>>(ei, E, eb, sm, hbuf, outb);
        bn_stats<<<FF, 256, 0, stream>>>(outb, mu, var);
        bn_relu<<<nfBlk, 256, 0, stream>>>(outb, mu, var, g, be, feat);
        in = feat;
    }

    // final linear: [N,256] @ [256,32] + bl  -> d_out  (TN=2, 1 col group)
    const int gemmBlk2 = stripGroups;
    gemm_ker<2><<<gemmBlk2, 256, 0, stream>>>(feat, (const float*)d_in[27],
                                              (const float*)d_in[28],
                                              (float*)d_out, NN, FF, 32);
}